// PointChain_8710193677069
// MI455X (gfx1250) — compile-verified
//
#include <hip/hip_runtime.h>
#include <math.h>

// Problem constants (match reference)
#define NBC     64
#define NPERC   131072
#define NPTS    (NBC * NPERC)      // 8388608 points
#define NTYPESC 16
#define NITERC  10
#define ACCC    1e-9f
#define BLK     256
#define NBLK    (NPTS / BLK)       // 32768 blocks
#define NWAVE   (BLK / 32)         // 8 waves per block
// NPERC % BLK == 0 -> chain boundaries always coincide with block boundaries.

typedef __attribute__((ext_vector_type(2))) float v2f;
typedef __attribute__((ext_vector_type(8))) float v8f;

// ---------------------------------------------------------------------------
// CDNA5 async global->LDS copy (ASYNCcnt-tracked DMA path) + wait.
// ---------------------------------------------------------------------------
__device__ __forceinline__ unsigned lds_off_u32(const void* p) {
  // addrspace(3) offset == low 32 bits of the generic shared-aperture address
  return (unsigned)(unsigned long long)p;
}

__device__ __forceinline__ void async_g2l_b32(unsigned lds_off,
                                              const float* gptr) {
  asm volatile("global_load_async_to_lds_b32 %0, %1, off"
               :: "v"(lds_off), "v"(gptr)
               : "memory");
}

__device__ __forceinline__ void wait_async0() {
  asm volatile("s_wait_asynccnt 0" ::: "memory");
}

// ---------------------------------------------------------------------------
// Deterministic full-wave (32 lane) sum using V_WMMA_F32_16X16X4_F32.
// A: lane value in VGPR0, 0 in VGPR1; B: ones -> D[m][n] = v(m)+v(m+16).
// Sum of the 8 D VGPRs per lane + one xor-16 shuffle = full 32-lane sum.
// ---------------------------------------------------------------------------
__device__ __forceinline__ float wave_sum32(float v) {
#if __has_builtin(__builtin_amdgcn_wmma_f32_16x16x4_f32)
  v2f a; a[0] = v;    a[1] = 0.0f;
  v2f b; b[0] = 1.0f; b[1] = 1.0f;
  v8f c = (v8f)0.0f;
  c = __builtin_amdgcn_wmma_f32_16x16x4_f32(false, a, false, b, (short)0, c,
                                            false, false);
  float s = ((c[0] + c[1]) + (c[2] + c[3])) + ((c[4] + c[5]) + (c[6] + c[7]));
  s += __shfl_xor(s, 16, 32);
  return s;
#else
  #pragma unroll
  for (int m = 16; m; m >>= 1) v += __shfl_xor(v, m, 32);
  return v;
#endif
}

// Deterministic 7-quantity block sum with a SINGLE barrier phase.
// sm must hold NWAVE*7 floats. Results valid in thread 0 only.
__device__ __forceinline__ void block_sum7(const float v[7], float* sm,
                                           float out[7]) {
  float w[7];
  #pragma unroll
  for (int k = 0; k < 7; ++k) w[k] = wave_sum32(v[k]);
  const int wave = threadIdx.x >> 5;
  const int lane = threadIdx.x & 31;
  if (lane == 0) {
    #pragma unroll
    for (int k = 0; k < 7; ++k) sm[wave * 7 + k] = w[k];
  }
  __syncthreads();
  if (threadIdx.x == 0) {
    #pragma unroll
    for (int k = 0; k < 7; ++k) out[k] = 0.0f;
    #pragma unroll
    for (int wv = 0; wv < NWAVE; ++wv) {
      #pragma unroll
      for (int k = 0; k < 7; ++k) out[k] += sm[wv * 7 + k];
    }
  }
}

__device__ __forceinline__ int clampi(int v, int lo, int hi) {
  return v < lo ? lo : (v > hi ? hi : v);
}

// ---------------------------------------------------------------------------
// Fused kernel: async-stage x[i0-1 .. i0+BLK+1] into LDS, compute
//   c_b = |dx_b|^2 - d_b^2,  lam_i = 2*(c_{i-1} dx_{i-1} - c_i dx_i),
// share lam through LDS (thread 0 additionally computes the lam halo for
// point i0+BLK from the staged tile), then per-bond p = dx.dlam,
// q = |dlam|^2 and block partials of:
//   sum|lam|^2, Scc, Scp, Scq, Spp, Spq, Sqq   (WMMA wave reductions)
// ---------------------------------------------------------------------------
__global__ void k_fused(const float* __restrict__ x, const float* __restrict__ d0,
                        const int* __restrict__ z, float* __restrict__ lam,
                        float* __restrict__ part_ll, float* __restrict__ part_m) {
  __shared__ float sx[3 * (BLK + 3)];   // slot s holds point (i0-1+s), clamped
  __shared__ float sl[3 * (BLK + 1)];   // slot t holds lam_{i0+t}, t in [0,BLK]
  __shared__ float sm[NWAVE * 7];
  const int t  = threadIdx.x;
  const int i0 = blockIdx.x * BLK;
  const int i  = i0 + t;

  // ---- stage tile via CDNA5 async DMA (3 dwords per lane + halo) ----------
  {
    const unsigned o = lds_off_u32(&sx[3 * (t + 1)]);
    const float* g = &x[3 * i];
    async_g2l_b32(o,     g);
    async_g2l_b32(o + 4, g + 1);
    async_g2l_b32(o + 8, g + 2);
    if (t < 3) {
      const int slot = (t == 0) ? 0 : (BLK + t);         // 0, BLK+1, BLK+2
      const int gidx = clampi(i0 - 1 + slot, 0, NPTS - 1);
      const unsigned oh = lds_off_u32(&sx[3 * slot]);
      const float* gh = &x[3 * gidx];
      async_g2l_b32(oh,     gh);
      async_g2l_b32(oh + 4, gh + 1);
      async_g2l_b32(oh + 8, gh + 2);
    }
  }
  wait_async0();     // own wave's async transfers landed in LDS
  __syncthreads();   // all waves' transfers visible

  // ---- per-point lam from the staged tile ---------------------------------
  const int  j  = i & (NPERC - 1);
  const bool hp = (j != 0);            // only possible at t==0
  const bool hn = (j != NPERC - 1);    // only possible at t==BLK-1
  const int  s  = t + 1;

  const float xi0 = sx[3 * s + 0],     xi1 = sx[3 * s + 1],     xi2 = sx[3 * s + 2];
  const float xp0 = sx[3 * (s-1) + 0], xp1 = sx[3 * (s-1) + 1], xp2 = sx[3 * (s-1) + 2];
  const float xn0 = sx[3 * (s+1) + 0], xn1 = sx[3 * (s+1) + 1], xn2 = sx[3 * (s+1) + 2];
  const int zi = z[i];
  const int zp = z[i - (hp ? 1 : 0)];
  const int zn = z[i + (hn ? 1 : 0)];
  const float dp = d0[zp * NTYPESC + zi];
  const float dn = d0[zi * NTYPESC + zn];

  const float ap0 = xi0 - xp0, ap1 = xi1 - xp1, ap2 = xi2 - xp2;  // bond i-1
  const float an0 = xn0 - xi0, an1 = xn1 - xi1, an2 = xn2 - xi2;  // bond i
  float cp = ap0 * ap0 + ap1 * ap1 + ap2 * ap2 - dp * dp;
  float cn = an0 * an0 + an1 * an1 + an2 * an2 - dn * dn;
  cp = hp ? cp : 0.0f;
  cn = hn ? cn : 0.0f;

  const float l0 = 2.0f * (cp * ap0 - cn * an0);
  const float l1 = 2.0f * (cp * ap1 - cn * an1);
  const float l2 = 2.0f * (cp * ap2 - cn * an2);
  lam[3 * i + 0] = l0;
  lam[3 * i + 1] = l1;
  lam[3 * i + 2] = l2;
  sl[3 * t + 0] = l0;
  sl[3 * t + 1] = l1;
  sl[3 * t + 2] = l2;

  // ---- thread 0: lam halo for point e = i0+BLK (slots BLK..BLK+2) ---------
  // Only consumed when bond (i0+BLK-1) is active, which implies e is interior
  // to the same chain; in all other cases the masked moments ignore it.
  if (t == 0) {
    const int e  = clampi(i0 + BLK, 0, NPTS - 1);
    const int je = e & (NPERC - 1);
    const bool ehp = (je != 0);
    const bool ehn = (je != NPERC - 1);
    const float ei0 = sx[3*(BLK+1)+0], ei1 = sx[3*(BLK+1)+1], ei2 = sx[3*(BLK+1)+2];
    const float ep0 = sx[3*BLK+0],     ep1 = sx[3*BLK+1],     ep2 = sx[3*BLK+2];
    const float en0 = sx[3*(BLK+2)+0], en1 = sx[3*(BLK+2)+1], en2 = sx[3*(BLK+2)+2];
    const int ezi = z[e];
    const int ezp = z[clampi(e - 1, 0, NPTS - 1)];
    const int ezn = z[clampi(e + 1, 0, NPTS - 1)];
    const float edp = d0[ezp * NTYPESC + ezi];
    const float edn = d0[ezi * NTYPESC + ezn];
    const float eap0 = ei0 - ep0, eap1 = ei1 - ep1, eap2 = ei2 - ep2;
    const float ean0 = en0 - ei0, ean1 = en1 - ei1, ean2 = en2 - ei2;
    float ecp = eap0*eap0 + eap1*eap1 + eap2*eap2 - edp*edp;
    float ecn = ean0*ean0 + ean1*ean1 + ean2*ean2 - edn*edn;
    ecp = ehp ? ecp : 0.0f;
    ecn = ehn ? ecn : 0.0f;
    sl[3*BLK+0] = 2.0f * (ecp * eap0 - ecn * ean0);
    sl[3*BLK+1] = 2.0f * (ecp * eap1 - ecn * ean1);
    sl[3*BLK+2] = 2.0f * (ecp * eap2 - ecn * ean2);
  }
  __syncthreads();

  // ---- per-bond moments (bond i, active iff hn) ---------------------------
  const float dl0 = sl[3*(t+1)+0] - l0;
  const float dl1 = sl[3*(t+1)+1] - l1;
  const float dl2 = sl[3*(t+1)+2] - l2;
  const float c = cn;  // already masked
  float p = an0 * dl0 + an1 * dl1 + an2 * dl2;
  float q = dl0 * dl0 + dl1 * dl1 + dl2 * dl2;
  p = hn ? p : 0.0f;
  q = hn ? q : 0.0f;

  const float ll = l0 * l0 + l1 * l1 + l2 * l2;
  const float vals[7] = {ll, c * c, c * p, c * q, p * p, p * q, q * q};
  float S[7];
  block_sum7(vals, sm, S);
  if (t == 0) {
    part_ll[blockIdx.x] = S[0];
    #pragma unroll
    for (int k = 0; k < 6; ++k) part_m[k * NBLK + blockIdx.x] = S[k + 1];
  }
}

// ---------------------------------------------------------------------------
// Single-block final reduction + exact scalar backtracking line search on
//   ctry(a) = Scc - 4a*Scp + a^2*(2*Scq + 4*Spp) - 4a^3*Spq + a^4*Sqq
// st[]: [0]=alpha (frozen once done), [1]=update scale this iter, [2]=done.
// ---------------------------------------------------------------------------
__global__ void k_reduce_ls(const float* __restrict__ part_ll,
                            const float* __restrict__ part_m,
                            float* __restrict__ st, int iter) {
  __shared__ float sm[NWAVE * 7];
  float acc[7] = {0.f, 0.f, 0.f, 0.f, 0.f, 0.f, 0.f};
  for (int k = threadIdx.x; k < NBLK; k += BLK) {
    acc[0] += part_ll[k];
    #pragma unroll
    for (int t = 0; t < 6; ++t) acc[t + 1] += part_m[t * NBLK + k];
  }
  float S[7];
  block_sum7(acc, sm, S);

  if (threadIdx.x == 0) {
    const float Sll = S[0], Scc = S[1], Scp = S[2], Scq = S[3];
    const float Spp = S[4], Spq = S[5], Sqq = S[6];
    const float alpha_prev = st[0];
    const bool  done_prev  = (st[2] != 0.0f);
    const float cnorm = Scc;

    float a = (iter == 0) ? (1.0f / sqrtf(Sll)) : alpha_prev;
    int   lsiter = 0;
    float ctry   = __builtin_inff();
    bool  fin    = false;
    while (!fin) {
      const float a2 = a * a;
      ctry = Scc - 4.0f * a * Scp + a2 * (2.0f * Scq + 4.0f * Spp)
             - 4.0f * a2 * a * Spq + a2 * a2 * Sqq;
      const bool ok = ctry < cnorm;
      if (!ok) { a *= 0.5f; ++lsiter; }
      fin = ok || (lsiter > 10);
    }
    if (lsiter == 0 && ctry > ACCC) a *= 1.5f;

    st[0] = done_prev ? alpha_prev : a;
    st[1] = done_prev ? 0.0f : a;
    st[2] = (done_prev || (ctry < ACCC)) ? 1.0f : 0.0f;
  }
}

// ---------------------------------------------------------------------------
// x -= upd * lam  (upd == 0 once converged -> x frozen exactly)
// ---------------------------------------------------------------------------
__global__ void k_upd(float* __restrict__ xo, const float* __restrict__ lam,
                      const float* __restrict__ st) {
  const int i = blockIdx.x * BLK + threadIdx.x;
  const float a = st[1];
  xo[3 * i + 0] -= a * lam[3 * i + 0];
  xo[3 * i + 1] -= a * lam[3 * i + 1];
  xo[3 * i + 2] -= a * lam[3 * i + 2];
}

__global__ void k_init(float* __restrict__ st) {
  st[0] = 0.0f; st[1] = 0.0f; st[2] = 0.0f; st[3] = 0.0f;
}

// ---------------------------------------------------------------------------
extern "C" void kernel_launch(void* const* d_in, const int* in_sizes, int n_in,
                              void* d_out, int out_size, void* d_ws,
                              size_t ws_size, hipStream_t stream) {
  const float* x  = (const float*)d_in[0];
  const float* d0 = (const float*)d_in[1];
  // d_in[2] = batch: unused (contiguous chains, NB static)
  const int*   z  = (const int*)d_in[3];
  float* xo = (float*)d_out;

  char*  w       = (char*)d_ws;
  float* lam     = (float*)w;                                      // NPTS*3 f32
  float* part_ll = (float*)(w + (size_t)NPTS * 3 * sizeof(float)); // NBLK
  float* part_m  = part_ll + NBLK;                                 // 6*NBLK
  float* st      = part_m + 6 * NBLK;                              // 4 scalars

  // Work in-place on d_out; never mutate inputs.
  hipMemcpyAsync(xo, x, (size_t)NPTS * 3 * sizeof(float),
                 hipMemcpyDeviceToDevice, stream);
  k_init<<<1, 1, 0, stream>>>(st);

  for (int it = 0; it < NITERC; ++it) {
    k_fused<<<NBLK, BLK, 0, stream>>>(xo, d0, z, lam, part_ll, part_m);
    k_reduce_ls<<<1, BLK, 0, stream>>>(part_ll, part_m, st, it);
    k_upd<<<NBLK, BLK, 0, stream>>>(xo, lam, st);
  }
}